// EdgeDecoderBilinear_9844065042711
// MI455X (gfx1250) — compile-verified
//
#include <hip/hip_runtime.h>

// EdgeDecoderBilinear: logits = E @ E^T * scale + bias ; sigmoid ; strict-upper threshold.
// f32 WMMA path (V_WMMA_F32_16X16X4_F32) keeps full precision; kernel is store-bound.

typedef __attribute__((ext_vector_type(2))) float v2f;
typedef __attribute__((ext_vector_type(8))) float v8f;

#define D_MODEL 128
#define N_NODES 2048
#define BATCH   8

__global__ __launch_bounds__(256)
void EdgeDecoderBilinear_kernel(const float* __restrict__ E,      // [B, N, d]
                                const float* __restrict__ scale,  // [1]
                                const float* __restrict__ bias,   // [1]
                                float* __restrict__ probs,        // [B, N, N]
                                float* __restrict__ adj)          // [B, N, N] (0/1)
{
    const int lane = threadIdx.x & 31;
    const int wave = threadIdx.x >> 5;

    const int jt = blockIdx.x;              // 0..127  (16 cols each)
    const int it = blockIdx.y * 8 + wave;   // 0..127  (16 rows each)
    const int b  = blockIdx.z;

    const int row0 = it * 16;
    const int col0 = jt * 16;

    const float s = scale[0];
    const float t = bias[0];

    // WMMA f32 16x16x4 operand layout (wave32):
    //   A: lane L -> M = L%16 ; VGPR pair holds K = 2*(L/16) and 2*(L/16)+1
    //   B: lane L -> N = L%16 ; same K split (B tile = E^T, so read E rows of the j-block)
    const int lm = lane & 15;            // M for A / N for B
    const int kh = (lane >> 4) << 1;     // 0 or 2

    const float* Eb   = E + (size_t)b * N_NODES * D_MODEL;
    const float* arow = Eb + (size_t)(row0 + lm) * D_MODEL + kh;
    const float* brow = Eb + (size_t)(col0 + lm) * D_MODEL + kh;

    v8f acc = {};
#pragma unroll
    for (int kk = 0; kk < D_MODEL; kk += 4) {
        v2f a  = *(const v2f*)(arow + kk);   // global_load_b64
        v2f bb = *(const v2f*)(brow + kk);   // global_load_b64
        // 8 args: (neg_a, A, neg_b, B, c_mod, C, reuse_a, reuse_b)
        acc = __builtin_amdgcn_wmma_f32_16x16x4_f32(
            /*neg_a=*/false, a, /*neg_b=*/false, bb,
            /*c_mod=*/(short)0, acc, /*reuse_a=*/false, /*reuse_b=*/false);
    }

    // D layout: VGPR v, lane L -> n = L%16, m = v + 8*(L/16)
    const int n      = col0 + lm;
    const int mbase  = row0 + ((lane >> 4) << 3);
    const size_t outBase = (size_t)b * N_NODES * N_NODES;

#pragma unroll
    for (int v = 0; v < 8; ++v) {
        const int m = mbase + v;
        const float logit = acc[v] * s + t;
        const float p = 1.0f / (1.0f + __expf(-logit));
        const size_t idx = outBase + (size_t)m * N_NODES + n;
        probs[idx] = p;
        adj[idx]   = (p > 0.5f && n > m) ? 1.0f : 0.0f;  // strict upper triangle
    }
}

extern "C" void kernel_launch(void* const* d_in, const int* in_sizes, int n_in,
                              void* d_out, int out_size, void* d_ws, size_t ws_size,
                              hipStream_t stream) {
    const float* E     = (const float*)d_in[0];   // node_embeds [8,2048,128]
    // d_in[1] = weight [128,128] — dead code in the reference math
    const float* scale = (const float*)d_in[2];
    const float* bias  = (const float*)d_in[3];

    float* probs = (float*)d_out;
    float* adj   = probs + (size_t)BATCH * N_NODES * N_NODES;

    dim3 block(256);                                   // 8 waves (wave32)
    dim3 grid(N_NODES / 16, N_NODES / (16 * 8), BATCH); // 128 x 16 x 8

    hipLaunchKernelGGL(EdgeDecoderBilinear_kernel, grid, block, 0, stream,
                       E, scale, bias, probs, adj);
}